// CLAHE_55336358642728
// MI455X (gfx1250) — compile-verified
//
#include <hip/hip_runtime.h>
#include <stdint.h>

#define H_ 4096
#define W_ 4096
#define HW ((size_t)H_ * (size_t)W_)
#define TH 512
#define TW 512
#define GY 8
#define GX 8
#define NT 64
#define AREA (TH * TW)
#define LIMIT 2048          /* max(int(2.0*AREA/256),1) */
#define EPS_T 0.008856f

typedef float v2f __attribute__((ext_vector_type(2)));
typedef float v8f __attribute__((ext_vector_type(8)));

__device__ __forceinline__ float clamp255(float v) {
    return fminf(fmaxf(v, 0.f), 255.f);
}

__device__ __forceinline__ void rgb_to_lab_u8(float r, float g, float b,
                                              float& L8v, float& a8v, float& b8v) {
    // u8 quantize like reference: floor(img*255) clipped, back to [0,1]
    float sr = clamp255(floorf(r * 255.f)) * (1.f / 255.f);
    float sg = clamp255(floorf(g * 255.f)) * (1.f / 255.f);
    float sb = clamp255(floorf(b * 255.f)) * (1.f / 255.f);
    auto lin = [](float s) {
        return s > 0.04045f ? powf((s + 0.055f) * (1.f / 1.055f), 2.4f)
                            : s * (1.f / 12.92f);
    };
    float lr = lin(sr), lg = lin(sg), lb = lin(sb);
    float X = 0.412453f * lr + 0.357580f * lg + 0.180423f * lb;
    float Y = 0.212671f * lr + 0.715160f * lg + 0.072169f * lb;
    float Z = 0.019334f * lr + 0.119193f * lg + 0.950227f * lb;
    float tx = X * (1.f / 0.950456f);
    float ty = Y;
    float tz = Z * (1.f / 1.088754f);
    auto fcv = [](float t) {
        return t > EPS_T ? cbrtf(t) : 7.787f * t + (16.f / 116.f);
    };
    float fx = fcv(tx), fy = fcv(ty), fz = fcv(tz);
    float L = (ty > EPS_T) ? (116.f * fy - 16.f) : (903.3f * ty);
    float a = 500.f * (fx - fy);
    float bb = 200.f * (fy - fz);
    L8v = clamp255(rintf(L * (255.f / 100.f)));
    a8v = clamp255(rintf(a + 128.f));
    b8v = clamp255(rintf(bb + 128.f));
}

// ---------------- Kernel 1: RGB->Lab + per-tile histograms ----------------
// grid (8, 1024), block 256. Each block: one tile column, 4 rows (512x4 px).
__global__ void k1_lab_hist(const float* __restrict__ img,
                            uint8_t* __restrict__ L8, uint8_t* __restrict__ A8,
                            uint8_t* __restrict__ B8, float* __restrict__ fstage,
                            int u8mode, uint32_t* __restrict__ hist) {
    __shared__ uint32_t lh[256];
    int tid = threadIdx.x;
    lh[tid] = 0u;
    __syncthreads();

    int tileX = blockIdx.x;
    int rowBase = blockIdx.y * 4;
    int tileY = rowBase >> 9;

    #pragma unroll
    for (int i = 0; i < 8; ++i) {
        int linear = tid + i * 256;        // 0..2047 within 512x4 patch
        int xl = linear & 511;
        int yl = linear >> 9;
        int x = tileX * 512 + xl;
        int y = rowBase + yl;
        size_t p = (size_t)y * W_ + (size_t)x;
        float r = img[p];
        float g = img[p + HW];
        float b = img[p + 2 * HW];
        float l8v, a8v, b8v;
        rgb_to_lab_u8(r, g, b, l8v, a8v, b8v);
        if (u8mode) {
            L8[p] = (uint8_t)l8v;
            A8[p] = (uint8_t)a8v;
            B8[p] = (uint8_t)b8v;
        } else {
            fstage[p] = l8v;
            fstage[p + HW] = a8v;
            fstage[p + 2 * HW] = b8v;
        }
        atomicAdd(&lh[(int)l8v], 1u);
    }
    __syncthreads();
    uint32_t c = lh[tid];
    if (c) atomicAdd(&hist[(tileY * 8 + tileX) * 256 + tid], c);
}

// ---------------- Kernel 2: clip/redistribute + WMMA cumsum -> LUT --------
// grid 64 (one tile per block), block 32 (one wave).
// Cumsum of 256 bins: reshape hist to 16x16 = H; intra-row inclusive prefix
// P = H x U (U upper-triangular ones) via 4 chained V_WMMA_F32_16X16X4_F32
// (f32 WMMA is exact for integer counts < 2^24); row offsets finish the scan.
__global__ void k2_lut(const uint32_t* __restrict__ hist,
                       uint8_t* __restrict__ lut) {
    __shared__ float histf[256];
    __shared__ float pre[256];
    __shared__ float rowoff[16];

    int tile = blockIdx.x;
    int lane = threadIdx.x;           // 0..31
    const uint32_t* h = hist + tile * 256;

    // excess over clip limit (wave reduction)
    float hv[8];
    float ex = 0.f;
    #pragma unroll
    for (int i = 0; i < 8; ++i) {
        float v = (float)h[lane + i * 32];
        hv[i] = v;
        ex += fmaxf(v - (float)LIMIT, 0.f);
    }
    #pragma unroll
    for (int off = 16; off >= 1; off >>= 1) ex += __shfl_xor(ex, off, 32);

    float batch = floorf(ex * (1.f / 256.f));
    int residual = (int)(ex - batch * 256.f);
    int stepd = residual > 1 ? residual : 1;
    int step = 256 / stepd;
    if (step < 1) step = 1;

    #pragma unroll
    for (int i = 0; i < 8; ++i) {
        int bin = lane + i * 32;
        float v = fminf(hv[i], (float)LIMIT) + batch;
        if ((bin % step) == 0 && (bin / step) < residual) v += 1.f;
        histf[bin] = v;
    }
    __syncthreads();

    // WMMA: P = H(16x16) x U(16x16 upper-tri ones), K split into 4 chunks of 4.
    // A (16x4 f32) layout: lanes 0-15 rows M=0..15 with K={4c+0,4c+1},
    //                      lanes 16-31 same rows with K={4c+2,4c+3}.
    // B (4x16 f32) mirrored: VGPR0 holds K={4c+0,4c+2}, VGPR1 K={4c+1,4c+3}.
    v8f acc = {};
    int half = lane >> 4;             // 0 or 1
    int m = lane & 15;                // row for A, column for B/D
    #pragma unroll
    for (int c = 0; c < 4; ++c) {
        int k0 = 4 * c + half * 2;
        int k1 = k0 + 1;
        v2f A, B;
        A.x = histf[m * 16 + k0];
        A.y = histf[m * 16 + k1];
        B.x = (k0 <= m) ? 1.f : 0.f;  // U[k][n] = (k <= n)
        B.y = (k1 <= m) ? 1.f : 0.f;
        acc = __builtin_amdgcn_wmma_f32_16x16x4_f32(
            false, A, false, B, (short)0, acc, false, false);
    }

    // D layout: VGPR r -> row (r + 8*half), col = lane&15
    #pragma unroll
    for (int r = 0; r < 8; ++r) {
        int row = r + half * 8;
        pre[row * 16 + m] = acc[r];
    }
    __syncthreads();

    if (lane < 16) {
        float s = 0.f;
        for (int j = 0; j < lane; ++j) s += pre[j * 16 + 15];
        rowoff[lane] = s;
    }
    __syncthreads();

    #pragma unroll
    for (int i = 0; i < 8; ++i) {
        int bin = lane + i * 32;
        float cum = pre[bin] + rowoff[bin >> 4];
        float l = clamp255(rintf(cum * (255.f / (float)AREA)));
        lut[tile * 256 + bin] = (uint8_t)l;
    }
}

// ---------------- Kernel 3: bilinear LUT + Lab->RGB -----------------------
// grid 4096 (one image row per block), block 256, 16 px/thread. LUT in LDS.
__global__ void k3_apply(const uint8_t* __restrict__ L8,
                         const uint8_t* __restrict__ A8,
                         const uint8_t* __restrict__ B8,
                         const float* __restrict__ fstage, int u8mode,
                         const uint8_t* __restrict__ lut,
                         float* __restrict__ out) {
    __shared__ uint8_t slut[NT * 256];
    {
        const uint4* src = (const uint4*)lut;
        uint4* dst = (uint4*)slut;
        #pragma unroll
        for (int i = 0; i < 4; ++i) dst[threadIdx.x + i * 256] = src[threadIdx.x + i * 256];
    }
    __syncthreads();

    int y = blockIdx.x;
    float tyf = (float)y * (1.f / (float)TH) - 0.5f;
    float ty1f = floorf(tyf);
    float ya = tyf - ty1f;
    int ty1 = (int)ty1f;
    int ty2 = min(ty1 + 1, GY - 1);
    ty1 = min(max(ty1, 0), GY - 1);

    #pragma unroll 4
    for (int i = 0; i < 16; ++i) {
        int x = threadIdx.x + i * 256;
        size_t p = (size_t)y * W_ + (size_t)x;

        float l8v, a8v, b8v;
        if (u8mode) {
            l8v = (float)L8[p];
            a8v = (float)A8[p];
            b8v = (float)B8[p];
        } else {
            l8v = fstage[p];
            a8v = fstage[p + HW];
            b8v = fstage[p + 2 * HW];
        }
        int v = (int)l8v;

        float txf = (float)x * (1.f / (float)TW) - 0.5f;
        float tx1f = floorf(txf);
        float xa = txf - tx1f;
        int tx1 = (int)tx1f;
        int tx2 = min(tx1 + 1, GX - 1);
        tx1 = min(max(tx1, 0), GX - 1);

        float l11 = (float)slut[(ty1 * 8 + tx1) * 256 + v];
        float l12 = (float)slut[(ty1 * 8 + tx2) * 256 + v];
        float l21 = (float)slut[(ty2 * 8 + tx1) * 256 + v];
        float l22 = (float)slut[(ty2 * 8 + tx2) * 256 + v];
        float res = (l11 * (1.f - xa) + l12 * xa) * (1.f - ya) +
                    (l21 * (1.f - xa) + l22 * xa) * ya;
        float cl = clamp255(rintf(res));

        // Lab -> RGB
        float L = cl * (100.f / 255.f);
        float a = a8v - 128.f;
        float b = b8v - 128.f;
        float fy = (L + 16.f) * (1.f / 116.f);
        float fx = fy + a * (1.f / 500.f);
        float fz = fy - b * (1.f / 200.f);
        auto finv = [](float f) {
            float f3 = f * f * f;
            return f3 > EPS_T ? f3 : (f - 16.f / 116.f) * (1.f / 7.787f);
        };
        float Y = (L > 903.3f * EPS_T) ? (fy * fy * fy) : (L * (1.f / 903.3f));
        float X = finv(fx) * 0.950456f;
        float Z = finv(fz) * 1.088754f;

        float lr = 3.240479f * X - 1.537150f * Y - 0.498535f * Z;
        float lg = -0.969256f * X + 1.875991f * Y + 0.041556f * Z;
        float lb = 0.055648f * X - 0.204043f * Y + 1.057311f * Z;
        auto enc = [](float c) {
            return c > 0.0031308f ? 1.055f * powf(fabsf(c), 1.f / 2.4f) - 0.055f
                                  : 12.92f * c;
        };
        float r8 = clamp255(rintf(enc(lr) * 255.f));
        float g8 = clamp255(rintf(enc(lg) * 255.f));
        float b8o = clamp255(rintf(enc(lb) * 255.f));

        out[p] = r8 * (1.f / 255.f);
        out[p + HW] = g8 * (1.f / 255.f);
        out[p + 2 * HW] = b8o * (1.f / 255.f);
    }
}

extern "C" void kernel_launch(void* const* d_in, const int* in_sizes, int n_in,
                              void* d_out, int out_size, void* d_ws, size_t ws_size,
                              hipStream_t stream) {
    const float* img = (const float*)d_in[0];
    float* out = (float*)d_out;
    uint8_t* ws = (uint8_t*)d_ws;

    uint32_t* hist = (uint32_t*)ws;                 // 64 KB
    uint8_t* lut = ws + 65536;                      // 16 KB
    const size_t stageOff = 131072;                 // aligned staging start
    const size_t needU8 = stageOff + 3ull * HW;     // ~48.1 MB

    int u8mode = (ws_size >= needU8) ? 1 : 0;
    uint8_t *L8 = nullptr, *A8 = nullptr, *B8 = nullptr;
    if (u8mode) {
        L8 = ws + stageOff;
        A8 = L8 + HW;
        B8 = A8 + HW;
    }

    hipMemsetAsync(hist, 0, NT * 256 * sizeof(uint32_t), stream);
    k1_lab_hist<<<dim3(8, 1024), 256, 0, stream>>>(img, L8, A8, B8, out, u8mode, hist);
    k2_lut<<<NT, 32, 0, stream>>>(hist, lut);
    k3_apply<<<H_, 256, 0, stream>>>(L8, A8, B8, out, u8mode, lut, out);
}